// MultiHeadSelfAttention_15822659518596
// MI455X (gfx1250) — compile-verified
//
#include <hip/hip_runtime.h>

// Problem constants (from reference): B=4, S=2048, D=512, H=8, DK=64
#define BB 4
#define SS 2048
#define DD 512
#define HH 8
#define DKK 64

typedef float v2f __attribute__((ext_vector_type(2)));
typedef float v8f __attribute__((ext_vector_type(8)));

#define WAVES_PER_BLOCK 8

// D = A(16x4 f32) * B(4x16 f32) + C(16x16 f32), full fp32 WMMA path.
__device__ __forceinline__ v8f wmma4(v2f a, v2f b, v8f c) {
    return __builtin_amdgcn_wmma_f32_16x16x4_f32(
        /*neg_a=*/false, a, /*neg_b=*/false, b,
        /*c_mod=*/(short)0, c, /*reuse_a=*/false, /*reuse_b=*/false);
}

// ---------------------------------------------------------------------------
// Kernel 1: QK projection.  QK[row, e] = sum_d x[row,d]*W_qkv[wr(e),d] + b_qkv
//   e<512: q (wr = n*192+c), e>=512: k (wr = n*192+64+c).  V is dead code.
// Wave tile 32(rows) x 64(cols): 8 wmma per K-step, 6 b64 loads.
// Waves: 256 row-tiles x 16 col-groups = 4096 -> 512 blocks.
// ---------------------------------------------------------------------------
__global__ __launch_bounds__(256) void qk_proj_kernel(
    const float* __restrict__ x,      // (B*S, 512)
    const float* __restrict__ Wqkv,   // (1536, 512)
    const float* __restrict__ bqkv,   // (1536)
    float* __restrict__ QK)           // (B*S, 1024)
{
    const int lane   = threadIdx.x & 31;
    const int wave   = threadIdx.x >> 5;
    const int waveId = blockIdx.x * WAVES_PER_BLOCK + wave;
    const int rowTile = waveId >> 4;   // 256
    const int colGrp  = waveId & 15;   // 16
    const int m = lane & 15;
    const int g = lane >> 4;

    const int row0 = rowTile * 32;
    const int e0   = colGrp * 64;

    const float* Ap[2];
    Ap[0] = x + (size_t)(row0 + m) * DD + 2 * g;
    Ap[1] = x + (size_t)(row0 + 16 + m) * DD + 2 * g;

    const float* Bp[4];
    float bias[4];
#pragma unroll
    for (int t = 0; t < 4; ++t) {
        int e = e0 + t * 16 + m;
        int wr;
        if (e < DD) { int n = e >> 6, c = e & 63; wr = n * 192 + c; }
        else        { int e2 = e - DD; int n = e2 >> 6, c = e2 & 63; wr = n * 192 + 64 + c; }
        Bp[t]   = Wqkv + (size_t)wr * DD + 2 * g;
        bias[t] = bqkv[wr];
    }

    v8f acc[2][4];
#pragma unroll
    for (int u = 0; u < 2; ++u)
#pragma unroll
        for (int t = 0; t < 4; ++t) acc[u][t] = 0.0f;

    for (int k = 0; k < DD; k += 4) {
        v2f a0 = *(const v2f*)(Ap[0] + k);
        v2f a1 = *(const v2f*)(Ap[1] + k);
#pragma unroll
        for (int t = 0; t < 4; ++t) {
            v2f bf = *(const v2f*)(Bp[t] + k);
            acc[0][t] = wmma4(a0, bf, acc[0][t]);
            acc[1][t] = wmma4(a1, bf, acc[1][t]);
        }
    }

#pragma unroll
    for (int u = 0; u < 2; ++u)
#pragma unroll
        for (int t = 0; t < 4; ++t) {
            int e = e0 + t * 16 + m;
#pragma unroll
            for (int r = 0; r < 8; ++r) {
                int row = row0 + u * 16 + r + 8 * g;
                QK[(size_t)row * 1024 + e] = acc[u][t][r] + bias[t];
            }
        }
}

// ---------------------------------------------------------------------------
// Kernel 2: fused matrix M (the (S,S,H) scores tensor never materializes).
//   M[b, d, n*64+c] = (1/8) * sum_j W_fc[d, j*8+n] * K[b, j, n*64+c]
// Block = one (b,n) pair x 256 d-rows; K-slice chunks of 128 j-rows staged
// into LDS with CDNA5 async global->LDS copies (ASYNCcnt) and shared by all
// 8 waves. Wave tile 32(d) x 64(c).
// Blocks: 32 pairs x 2 d-groups = 64 x 256 threads.
// ---------------------------------------------------------------------------
__global__ __launch_bounds__(256) void build_m_kernel(
    const float* __restrict__ QK,    // (B*S, 1024); K half at col 512+
    const float* __restrict__ Wfc,   // (512, 16384)
    float* __restrict__ Mbuf)        // (B, 512 d, 512 nc)
{
    __shared__ float tileK[128 * 64];   // 32 KB: K[jc..jc+127][c0..c63]

    const int tid  = threadIdx.x;
    const int lane = tid & 31;
    const int wave = tid >> 5;
    const int pair   = blockIdx.x >> 1;   // 32: (b, n)
    const int dGroup = blockIdx.x & 1;    // 2
    const int b = pair >> 3;
    const int n = pair & 7;
    const int m = lane & 15;
    const int g = lane >> 4;

    const int d0 = dGroup * 256 + wave * 32;

    // A'[d][j] = Wfc[d, j*8 + n] (stride-8 over j)
    const float* Ap0 = Wfc + (size_t)(d0 + m) * (SS * HH) + n;
    const float* Ap1 = Wfc + (size_t)(d0 + 16 + m) * (SS * HH) + n;

    const float* gKbase = QK + (size_t)b * SS * 1024 + DD + n * DKK;

    const unsigned ldsBase = (unsigned)(unsigned long long)&tileK[0];
    const int jl = tid >> 4;          // 0..15 (row within 16-row copy slab)
    const int c4 = (tid & 15) * 4;    // 0..60 (b128 = 4 floats)

    v8f acc[2][4];
#pragma unroll
    for (int u = 0; u < 2; ++u)
#pragma unroll
        for (int t = 0; t < 4; ++t) acc[u][t] = 0.0f;

    for (int jc = 0; jc < SS; jc += 128) {
        // ---- stage K[jc..jc+127][0..63] into LDS via async copy ----
        const float* src0 = gKbase + (size_t)(jc + jl) * 1024 + c4;
#pragma unroll
        for (int i = 0; i < 8; ++i) {
            const float* src = src0 + (size_t)(i * 16) * 1024;
            unsigned dst = ldsBase + (unsigned)(((i * 16 + jl) * 64 + c4) * 4);
            asm volatile("global_load_async_to_lds_b128 %0, %1, off"
                         :: "v"(dst), "v"(src) : "memory");
        }
        asm volatile("s_wait_asynccnt 0x0" ::: "memory");
        __syncthreads();

        // ---- consume chunk: 8 wmma per K-step ----
        for (int k = 0; k < 128; k += 4) {
            const int j0 = k + 2 * g;
            const int jg = jc + j0;
            v2f a0, a1;
            a0.x = Ap0[(size_t)jg * HH];
            a0.y = Ap0[(size_t)(jg + 1) * HH];
            a1.x = Ap1[(size_t)jg * HH];
            a1.y = Ap1[(size_t)(jg + 1) * HH];
            const float* Lr = &tileK[j0 * 64];
#pragma unroll
            for (int t = 0; t < 4; ++t) {
                int c = t * 16 + m;
                v2f bf;
                bf.x = Lr[c];
                bf.y = Lr[64 + c];
                acc[0][t] = wmma4(a0, bf, acc[0][t]);
                acc[1][t] = wmma4(a1, bf, acc[1][t]);
            }
        }
        __syncthreads();   // all waves done reading before next chunk lands
    }

    const float scale = 0.125f;   // 1/sqrt(DK)
    float* Mout = Mbuf + ((size_t)b * DD + d0) * DD + n * DKK;
#pragma unroll
    for (int u = 0; u < 2; ++u)
#pragma unroll
        for (int t = 0; t < 4; ++t) {
            int c = t * 16 + m;
#pragma unroll
            for (int r = 0; r < 8; ++r) {
                int dl = u * 16 + r + 8 * g;
                Mout[(size_t)dl * DD + c] = acc[u][t][r] * scale;
            }
        }
}

// ---------------------------------------------------------------------------
// Kernel 3: out[b,i,d] = sum_nc Q[b,i,nc] * M[b,d,nc] + b_fc[d]
// Wave tile 32(i) x 64(d): 8 wmma per K-step, 6 b64 loads.
// Waves: 4 b x 64 i-tiles x 8 d-groups = 2048 -> 256 blocks.
// ---------------------------------------------------------------------------
__global__ __launch_bounds__(256) void out_gemm_kernel(
    const float* __restrict__ QK,    // Q half at cols 0..511
    const float* __restrict__ Mbuf,  // (B, 512, 512)
    const float* __restrict__ bfc,   // (512)
    float* __restrict__ out)         // (B*S, 512)
{
    const int lane   = threadIdx.x & 31;
    const int wave   = threadIdx.x >> 5;
    const int waveId = blockIdx.x * WAVES_PER_BLOCK + wave;
    const int b    = waveId >> 9;     // 512 waves per batch
    const int rem  = waveId & 511;
    const int iTile = rem >> 3;       // 64
    const int dGrp  = rem & 7;        // 8
    const int m = lane & 15;
    const int g = lane >> 4;

    const int i0 = iTile * 32;
    const int d0 = dGrp * 64;

    const float* Ap[2];
    Ap[0] = QK + (size_t)(b * SS + i0 + m) * 1024 + 2 * g;
    Ap[1] = QK + (size_t)(b * SS + i0 + 16 + m) * 1024 + 2 * g;

    const float* Bp[4];
#pragma unroll
    for (int t = 0; t < 4; ++t) {
        int d = d0 + t * 16 + m;
        Bp[t] = Mbuf + ((size_t)b * DD + d) * DD + 2 * g;
    }

    v8f acc[2][4];
#pragma unroll
    for (int u = 0; u < 2; ++u)
#pragma unroll
        for (int t = 0; t < 4; ++t) acc[u][t] = 0.0f;

    for (int k = 0; k < DD; k += 4) {
        v2f a0 = *(const v2f*)(Ap[0] + k);
        v2f a1 = *(const v2f*)(Ap[1] + k);
#pragma unroll
        for (int t = 0; t < 4; ++t) {
            v2f bf = *(const v2f*)(Bp[t] + k);
            acc[0][t] = wmma4(a0, bf, acc[0][t]);
            acc[1][t] = wmma4(a1, bf, acc[1][t]);
        }
    }

#pragma unroll
    for (int u = 0; u < 2; ++u)
#pragma unroll
        for (int t = 0; t < 4; ++t) {
            int d = d0 + t * 16 + m;
            float bias = bfc[d];
#pragma unroll
            for (int r = 0; r < 8; ++r) {
                int row = b * SS + i0 + u * 16 + r + 8 * g;
                out[(size_t)row * DD + d] = acc[u][t][r] + bias;
            }
        }
}

extern "C" void kernel_launch(void* const* d_in, const int* in_sizes, int n_in,
                              void* d_out, int out_size, void* d_ws, size_t ws_size,
                              hipStream_t stream) {
    (void)in_sizes; (void)n_in; (void)out_size; (void)ws_size;
    const float* x    = (const float*)d_in[0];
    const float* Wqkv = (const float*)d_in[1];
    const float* bqkv = (const float*)d_in[2];
    const float* Wfc  = (const float*)d_in[3];
    const float* bfc  = (const float*)d_in[4];
    float* out = (float*)d_out;

    // Workspace: QK (B*S x 1024 f32 = 33.5 MB), then M (B x 512 x 512 f32 = 4 MB)
    float* QK   = (float*)d_ws;
    float* Mbuf = QK + (size_t)(BB * SS) * 1024;

    qk_proj_kernel<<<512, 256, 0, stream>>>(x, Wqkv, bqkv, QK);
    build_m_kernel<<<64, 256, 0, stream>>>(QK, Wfc, Mbuf);
    out_gemm_kernel<<<256, 256, 0, stream>>>(QK, Mbuf, bfc, out);
}